// RefNet_Val_73126113182378
// MI455X (gfx1250) — compile-verified
//
#include <hip/hip_runtime.h>
#include <hip/hip_bf16.h>

typedef __attribute__((ext_vector_type(2))) float v2f;
typedef __attribute__((ext_vector_type(8))) float v8f;

#define S_SZ   20000
#define Q_SZ   32
#define N_SZ   200000
#define T_SZ   30
#define G_SZ   64
#define DIN    300
#define H_SZ   256
#define DSV    64
#define DLANG  64
#define DFUSE  128
#define DCAT   131   // DSV + 3 + DLANG
#define NQ     (N_SZ * Q_SZ)

// ---------------------------------------------------------------------------
// ws layout (floats):
//   gi_all  : 960*768      = 737280
//   sv_proj : 20000*128    = 2560000
//   lp      : 32*128       = 4096   (lang_proj + fuse_b folded in)
//   gsum    : 32*64        = 2048   (float)
//   gcnt    : 64  (int)
//   best    : 32  (int)
//   inter   : 32  (int)
//   uni     : 32  (int)
// ---------------------------------------------------------------------------

__global__ __launch_bounds__(256) void k_init(float* gsum) {
  // zero gsum(2048) + gcnt(64) + best(32) + inter(32) + uni(32) = 2208 words
  int* p = (int*)gsum;
  for (int i = threadIdx.x; i < 2208; i += 256) p[i] = 0;
}

// ---------------------------------------------------------------------------
// Kernel 1: gi_all = lang_feat(960,300) @ w_ih^T (300,768) + b_ih
// one wave per 16x16 C tile; tiles = 60 x 48 = 2880; 8 waves/block -> 360 blocks
// ---------------------------------------------------------------------------
__global__ __launch_bounds__(256) void k_gi(const float* __restrict__ x,
                                            const float* __restrict__ wih,
                                            const float* __restrict__ bih,
                                            float* __restrict__ gi) {
  int tid  = threadIdx.x;
  int lane = tid & 31;
  int task = blockIdx.x * 8 + (tid >> 5);   // 0..2879
  int mt = task / 48, nt = task % 48;
  int lrow = lane & 15, lhalf = lane >> 4, koff = lhalf * 2;
  int arow = mt * 16 + lrow;
  int ncol = nt * 16 + lrow;
  v8f acc = {};
  for (int k = 0; k < DIN; k += 4) {
    v2f a, b;
    a.x = x[arow * DIN + k + koff];
    a.y = x[arow * DIN + k + koff + 1];
    b.x = wih[ncol * DIN + k + koff];
    b.y = wih[ncol * DIN + k + koff + 1];
    acc = __builtin_amdgcn_wmma_f32_16x16x4_f32(false, a, false, b,
                                                (short)0, acc, false, false);
  }
  for (int i = 0; i < 8; ++i) {
    int m = mt * 16 + i + 8 * lhalf;
    int n = nt * 16 + lrow;
    gi[m * 768 + n] = acc[i] + bih[n];
  }
}

// ---------------------------------------------------------------------------
// Kernel 2: sequential GRU over T=30 (h in LDS) + squeeze MLP + lang_proj.
// 1 block, 1024 threads = 32 waves. Wave w owns the 16x16 (q,j) patch
// (mt = w>>4, jt = w&15) and computes its r/z/n gate tiles of gh = h@w_hh^T.
// ---------------------------------------------------------------------------
__global__ __launch_bounds__(1024) void k_gru(const float* __restrict__ gi,
                                              const float* __restrict__ whh,
                                              const float* __restrict__ bhh,
                                              const int*   __restrict__ lens,
                                              const float* __restrict__ sw1,
                                              const float* __restrict__ sb1,
                                              const float* __restrict__ sw2,
                                              const float* __restrict__ sb2,
                                              const float* __restrict__ fw,
                                              const float* __restrict__ fb,
                                              float* __restrict__ lp) {
  __shared__ float hs[Q_SZ][H_SZ];       // 32 KB
  __shared__ float l1s[Q_SZ][DLANG];     // 8 KB
  __shared__ float langs[Q_SZ][DLANG];   // 8 KB
  __shared__ int   lenls[Q_SZ];

  int tid  = threadIdx.x;
  int lane = tid & 31, wave = tid >> 5;
  for (int i = tid; i < Q_SZ * H_SZ; i += 1024) ((float*)hs)[i] = 0.f;
  if (tid < Q_SZ) lenls[tid] = lens[tid];
  __syncthreads();

  int mt = wave >> 4;               // 0..1  (q tile)
  int jt = wave & 15;               // 0..15 (j tile within 256)
  int lrow = lane & 15, lhalf = lane >> 4, koff = lhalf * 2;
  int arow = mt * 16 + lrow;
  int ncr  = jt * 16 + lrow;        // r-gate output column

  for (int t = 0; t < T_SZ; ++t) {
    v8f aR = {}, aZ = {}, aN = {};
    for (int k = 0; k < H_SZ; k += 4) {
      v2f a, br, bz, bn;
      a.x  = hs[arow][k + koff];
      a.y  = hs[arow][k + koff + 1];
      br.x = whh[ ncr        * H_SZ + k + koff];
      br.y = whh[ ncr        * H_SZ + k + koff + 1];
      bz.x = whh[(ncr + 256) * H_SZ + k + koff];
      bz.y = whh[(ncr + 256) * H_SZ + k + koff + 1];
      bn.x = whh[(ncr + 512) * H_SZ + k + koff];
      bn.y = whh[(ncr + 512) * H_SZ + k + koff + 1];
      aR = __builtin_amdgcn_wmma_f32_16x16x4_f32(false, a, false, br, (short)0, aR, false, false);
      aZ = __builtin_amdgcn_wmma_f32_16x16x4_f32(false, a, false, bz, (short)0, aZ, false, false);
      aN = __builtin_amdgcn_wmma_f32_16x16x4_f32(false, a, false, bn, (short)0, aN, false, false);
    }
    __syncthreads();   // all reads of hs done
    for (int i = 0; i < 8; ++i) {
      int q = mt * 16 + i + 8 * lhalf;          // 0..31
      int j = jt * 16 + lrow;                   // 0..255
      const float* gir = gi + (q * T_SZ + t) * 768;
      float ghr = aR[i] + bhh[j];
      float ghz = aZ[i] + bhh[256 + j];
      float ghn = aN[i] + bhh[512 + j];
      float rg  = 1.f / (1.f + __expf(-(gir[j] + ghr)));
      float zg  = 1.f / (1.f + __expf(-(gir[256 + j] + ghz)));
      float ng  = tanhf(gir[512 + j] + rg * ghn);
      float hold = hs[q][j];
      float hnew = (1.f - zg) * ng + zg * hold;
      hs[q][j] = (t < lenls[q]) ? hnew : hold;
    }
    __syncthreads();   // hs fully updated before next step's reads
  }

  // l1 = relu(h @ sqz_w1^T + b1)
  for (int idx = tid; idx < Q_SZ * DLANG; idx += 1024) {
    int q = idx >> 6, o = idx & 63;
    float a = sb1[o];
    for (int k = 0; k < H_SZ; ++k) a += hs[q][k] * sw1[o * H_SZ + k];
    l1s[q][o] = fmaxf(a, 0.f);
  }
  __syncthreads();
  // lang = l1 @ sqz_w2^T + b2
  for (int idx = tid; idx < Q_SZ * DLANG; idx += 1024) {
    int q = idx >> 6, o = idx & 63;
    float a = sb2[o];
    for (int k = 0; k < DLANG; ++k) a += l1s[q][k] * sw2[o * DLANG + k];
    langs[q][o] = a;
  }
  __syncthreads();
  // lp = lang @ fuse_w[:, 67:]^T + fuse_b   (fuse_b folded in here)
  for (int idx = tid; idx < Q_SZ * DFUSE; idx += 1024) {
    int q = idx >> 7, f = idx & 127;
    float a = fb[f];
    for (int k = 0; k < DLANG; ++k) a += langs[q][k] * fw[f * DCAT + 67 + k];
    lp[q * DFUSE + f] = a;
  }
}

// ---------------------------------------------------------------------------
// Kernel 3: sv_proj = [supvox_feat | coords] (20000,67) @ fuse_w[:, :67]^T
// one wave per 16x16 C tile; tiles = 1250 x 8; 8 waves/block -> 1250 blocks
// ---------------------------------------------------------------------------
__global__ __launch_bounds__(256) void k_svp(const float* __restrict__ feat,
                                             const float* __restrict__ coords,
                                             const float* __restrict__ fw,
                                             float* __restrict__ svp) {
  int tid  = threadIdx.x;
  int lane = tid & 31;
  int task = blockIdx.x * 8 + (tid >> 5);   // 0..9999
  int mt = task >> 3, nt = task & 7;
  int lrow = lane & 15, lhalf = lane >> 4, koff = lhalf * 2;
  int arow = mt * 16 + lrow;
  int ncol = nt * 16 + lrow;
  v8f acc = {};
  for (int k = 0; k < DSV; k += 4) {
    v2f a, b;
    a.x = feat[arow * DSV + k + koff];
    a.y = feat[arow * DSV + k + koff + 1];
    b.x = fw[ncol * DCAT + k + koff];
    b.y = fw[ncol * DCAT + k + koff + 1];
    acc = __builtin_amdgcn_wmma_f32_16x16x4_f32(false, a, false, b,
                                                (short)0, acc, false, false);
  }
  { // tail: columns 64..67 -> coords x,y,z and a zero pad column
    int c0 = DSV + koff;   // 64 or 66
    v2f a, b;
    a.x = (c0     < 67) ? coords[arow * 3 + (c0 - 64)] : 0.f;
    a.y = (c0 + 1 < 67) ? coords[arow * 3 + (c0 - 63)] : 0.f;
    b.x = (c0     < 67) ? fw[ncol * DCAT + c0]         : 0.f;
    b.y = (c0 + 1 < 67) ? fw[ncol * DCAT + c0 + 1]     : 0.f;
    acc = __builtin_amdgcn_wmma_f32_16x16x4_f32(false, a, false, b,
                                                (short)0, acc, false, false);
  }
  for (int i = 0; i < 8; ++i) {
    int m = mt * 16 + i + 8 * lhalf;
    int n = nt * 16 + lrow;
    svp[m * DFUSE + n] = acc[i];
  }
}

// ---------------------------------------------------------------------------
// Kernel 4: obj_score[s,q] = sigmoid( bd + sum_f relu(svp[s,f]+lp[q,f]) * wd[f] )
// accumulated into gsum[q, grps[s]]; also counts gcnt[g].
// 64 s-values per block of 256 threads (8 s x 32 q at a time).
// ---------------------------------------------------------------------------
__global__ __launch_bounds__(256) void k_score(const float* __restrict__ svp,
                                               const float* __restrict__ lp,
                                               const float* __restrict__ hw,
                                               const float* __restrict__ hb,
                                               const int*   __restrict__ grps,
                                               float* __restrict__ gsum,
                                               int*   __restrict__ gcnt) {
  __shared__ float lps[Q_SZ * DFUSE];   // 16 KB
  __shared__ float wds[DFUSE];
  __shared__ float rows[8][DFUSE];      // 4 KB
  int tid = threadIdx.x;
  for (int i = tid; i < Q_SZ * DFUSE; i += 256) lps[i] = lp[i];
  if (tid < DFUSE) wds[tid] = hw[DFUSE + tid] - hw[tid];
  __syncthreads();
  float bd = hb[1] - hb[0];
  int q = tid & 31, si = tid >> 5;
  int base = blockIdx.x * 64;
  for (int it = 0; it < 8; ++it) {
    int s0 = base + it * 8;
    __syncthreads();
    for (int i = tid; i < 8 * DFUSE; i += 256) {
      int s = s0 + (i >> 7);
      ((float*)rows)[i] = (s < S_SZ) ? svp[s * DFUSE + (i & 127)] : 0.f;
    }
    __syncthreads();
    int s = s0 + si;
    if (s < S_SZ) {
      float acc = bd;
      const float* lq = lps + q * DFUSE;
      for (int f = 0; f < DFUSE; ++f)
        acc += fmaxf(rows[si][f] + lq[f], 0.f) * wds[f];
      float sc = 1.f / (1.f + __expf(-acc));   // softmax(2)[1] = sigmoid(l1-l0)
      int g = grps[s];
      atomicAdd(&gsum[q * G_SZ + g], sc);
      if (q == 0) atomicAdd(&gcnt[g], 1);
    }
  }
}

// ---------------------------------------------------------------------------
// Kernel 5: best[q] = argmax_g gsum[q,g] / gcnt[g]   (first max, like jnp.argmax)
// ---------------------------------------------------------------------------
__global__ void k_best(const float* __restrict__ gsum,
                       const int* __restrict__ gcnt,
                       int* __restrict__ best) {
  int q = threadIdx.x;
  if (q < Q_SZ) {
    float bv = -3.0e38f; int bg = 0;
    for (int g = 0; g < G_SZ; ++g) {
      float v = gsum[q * G_SZ + g] / (float)gcnt[g];
      if (v > bv) { bv = v; bg = g; }
    }
    best[q] = bg;
  }
}

// ---------------------------------------------------------------------------
// Kernel 6: obj_pts[n,q] = (grps[supervox[n]] == best[q]); stream to d_out as
// float, reduce inter/union per q with integer atomics (deterministic).
// ---------------------------------------------------------------------------
__global__ __launch_bounds__(256) void k_iou(const int* __restrict__ supervox,
                                             const int* __restrict__ grps,
                                             const int* __restrict__ gt,
                                             const int* __restrict__ best,
                                             float* __restrict__ out,
                                             int* __restrict__ interA,
                                             int* __restrict__ uniA,
                                             int nthreads) {
  __shared__ int bests[Q_SZ];
  __shared__ int li_s[Q_SZ], lu_s[Q_SZ];
  int tid = threadIdx.x;
  if (tid < Q_SZ) { bests[tid] = best[tid]; li_s[tid] = 0; lu_s[tid] = 0; }
  __syncthreads();
  int idx0 = blockIdx.x * 256 + tid;
  int q = idx0 & 31;           // nthreads is a multiple of 32 -> q fixed
  int bq = bests[q];
  int li = 0, lu = 0;
  for (int idx = idx0; idx < NQ; idx += nthreads) {
    int n   = idx >> 5;
    int g   = grps[supervox[n]];
    int obj = (g == bq) ? 1 : 0;
    int gv  = gt[idx];                 // (N,Q) row-major: idx == n*32+q
    out[idx] = (float)obj;
    li += obj & gv;
    lu += obj | gv;
  }
  atomicAdd(&li_s[q], li);
  atomicAdd(&lu_s[q], lu);
  __syncthreads();
  if (tid < Q_SZ) {
    atomicAdd(&interA[tid], li_s[tid]);
    atomicAdd(&uniA[tid],  lu_s[tid]);
  }
}

__global__ void k_ious(const int* __restrict__ interA,
                       const int* __restrict__ uniA,
                       float* __restrict__ out) {
  int q = threadIdx.x;
  if (q < Q_SZ)
    out[NQ + q] = (float)interA[q] / ((float)uniA[q] + 1e-5f);
}

// ---------------------------------------------------------------------------
extern "C" void kernel_launch(void* const* d_in, const int* in_sizes, int n_in,
                              void* d_out, int out_size, void* d_ws, size_t ws_size,
                              hipStream_t stream) {
  (void)in_sizes; (void)n_in; (void)out_size; (void)ws_size;
  const float* lang_feat = (const float*)d_in[0];
  const float* sv_feat   = (const float*)d_in[1];
  const float* sv_coords = (const float*)d_in[2];
  const int*   lang_len  = (const int*)  d_in[3];
  const int*   grps      = (const int*)  d_in[4];
  const int*   supervox  = (const int*)  d_in[5];
  const int*   gt        = (const int*)  d_in[6];
  const float* w_ih      = (const float*)d_in[7];
  const float* w_hh      = (const float*)d_in[8];
  const float* b_ih      = (const float*)d_in[9];
  const float* b_hh      = (const float*)d_in[10];
  const float* sqz_w1    = (const float*)d_in[11];
  const float* sqz_b1    = (const float*)d_in[12];
  const float* sqz_w2    = (const float*)d_in[13];
  const float* sqz_b2    = (const float*)d_in[14];
  const float* fuse_w    = (const float*)d_in[15];
  const float* fuse_b    = (const float*)d_in[16];
  const float* heat_w    = (const float*)d_in[17];
  const float* heat_b    = (const float*)d_in[18];
  float* out = (float*)d_out;

  float* ws   = (float*)d_ws;
  float* gi   = ws;                       // 737280
  float* svp  = gi  + 960 * 768;          // 2560000
  float* lp   = svp + S_SZ * DFUSE;       // 4096
  float* gsum = lp  + Q_SZ * DFUSE;       // 2048
  int*   gcnt = (int*)(gsum + Q_SZ * G_SZ);
  int*   best = gcnt + G_SZ;
  int*   interA = best + Q_SZ;
  int*   uniA   = interA + Q_SZ;

  k_init<<<1, 256, 0, stream>>>(gsum);
  k_gi  <<<360, 256, 0, stream>>>(lang_feat, w_ih, b_ih, gi);
  k_gru <<<1, 1024, 0, stream>>>(gi, w_hh, b_hh, lang_len,
                                 sqz_w1, sqz_b1, sqz_w2, sqz_b2,
                                 fuse_w, fuse_b, lp);
  k_svp <<<1250, 256, 0, stream>>>(sv_feat, sv_coords, fuse_w, svp);
  k_score<<<(S_SZ + 63) / 64, 256, 0, stream>>>(svp, lp, heat_w, heat_b,
                                                grps, gsum, gcnt);
  k_best<<<1, 32, 0, stream>>>(gsum, gcnt, best);
  int blocks = 784;
  k_iou <<<blocks, 256, 0, stream>>>(supervox, grps, gt, best, out,
                                     interA, uniA, blocks * 256);
  k_ious<<<1, 32, 0, stream>>>(interA, uniA, out);
}